// SoftDomainAdaptiveReconstructor_3118146257161
// MI455X (gfx1250) — compile-verified
//
#include <hip/hip_runtime.h>
#include <hip/hip_bf16.h>
#include <math.h>

typedef __attribute__((ext_vector_type(2))) float v2f;
typedef __attribute__((ext_vector_type(4))) float v4f;
typedef __attribute__((ext_vector_type(8))) float v8f;

// Problem dims fixed by setup_inputs()
constexpr int B_ = 4, T_ = 8, S_ = 1024, P_ = 8192, D_ = 256;
constexpr int BT_ = B_ * T_;          // 32
constexpr int M_  = BT_ * S_;         // 32768 GEMM rows
constexpr int TOPK = 32;
constexpr float SIGMA_ = 0.05f;

#if __has_builtin(__builtin_amdgcn_wmma_f32_16x16x4_f32)
#define USE_WMMA_F32 1
#else
#define USE_WMMA_F32 0
#endif

// ---------------------------------------------------------------------------
// Kernel A: sensor_proj[m, n] = z[m, :] @ W_lat[:, n] + b_lat[n]   (fp32 WMMA)
// One wave -> 16(M) x 64(N) tile; K stepped by 4 via V_WMMA_F32_16X16X4_F32.
// fp32 A 16x4 frag: lanes 0-15 hold (K, K+1), lanes 16-31 hold (K+2, K+3).
// fp32 C/D: VGPR i -> M=i (lanes 0-15) / M=i+8 (lanes 16-31), N = lane&15.
// ---------------------------------------------------------------------------
#if USE_WMMA_F32
__global__ __launch_bounds__(256) void k_proj_gemm(
    const float* __restrict__ z, const float* __restrict__ W,
    const float* __restrict__ bias, float* __restrict__ proj)
{
  const int lane = threadIdx.x & 31;
  const int wave = blockIdx.x * (blockDim.x >> 5) + (threadIdx.x >> 5);
  const int mt = wave >> 2;            // 2048 M-tiles
  const int nt = wave & 3;             // 4 N-tiles of 64
  const int m0 = mt * 16;
  const int n0 = nt * 64;
  const int rowA = m0 + (lane & 15);
  const int kh   = (lane >> 4) * 2;    // 0 for lanes 0-15, 2 for lanes 16-31
  const int colL = lane & 15;

  __builtin_prefetch(&z[(size_t)rowA * D_], 0, 3);  // global_prefetch_b8

  v8f c0 = {}, c1 = {}, c2 = {}, c3 = {};

  for (int k = 0; k < D_; k += 4) {
    // A fragment (2 VGPRs): two consecutive K values per lane
    v2f a = *(const v2f*)&z[(size_t)rowA * D_ + k + kh];
    // B fragments: b.x = row (k+kh), b.y = row (k+kh+1), col = n0 + sub*16 + colL
    const float* wr0 = &W[(size_t)(k + kh) * D_ + n0 + colL];
    const float* wr1 = wr0 + D_;
    v2f b0 = { wr0[0],  wr1[0]  };
    v2f b1 = { wr0[16], wr1[16] };
    v2f b2 = { wr0[32], wr1[32] };
    v2f b3 = { wr0[48], wr1[48] };
    c0 = __builtin_amdgcn_wmma_f32_16x16x4_f32(false, a, false, b0, (short)0, c0, false, false);
    c1 = __builtin_amdgcn_wmma_f32_16x16x4_f32(false, a, false, b1, (short)0, c1, false, false);
    c2 = __builtin_amdgcn_wmma_f32_16x16x4_f32(false, a, false, b2, (short)0, c2, false, false);
    c3 = __builtin_amdgcn_wmma_f32_16x16x4_f32(false, a, false, b3, (short)0, c3, false, false);
  }

  const int rshift = (lane >> 4) * 8;
  const float bs0 = bias[n0 +  0 + colL];
  const float bs1 = bias[n0 + 16 + colL];
  const float bs2 = bias[n0 + 32 + colL];
  const float bs3 = bias[n0 + 48 + colL];
#pragma unroll
  for (int i = 0; i < 8; ++i) {
    const size_t r = (size_t)(m0 + i + rshift) * D_;
    proj[r + n0 +  0 + colL] = c0[i] + bs0;
    proj[r + n0 + 16 + colL] = c1[i] + bs1;
    proj[r + n0 + 32 + colL] = c2[i] + bs2;
    proj[r + n0 + 48 + colL] = c3[i] + bs3;
  }
}
#else
// Fallback (should not trigger per probe data): plain FMA GEMM
__global__ __launch_bounds__(256) void k_proj_gemm(
    const float* __restrict__ z, const float* __restrict__ W,
    const float* __restrict__ bias, float* __restrict__ proj)
{
  size_t gid = (size_t)blockIdx.x * blockDim.x + threadIdx.x;
  size_t stride = (size_t)gridDim.x * blockDim.x;
  for (size_t o = gid; o < (size_t)M_ * D_; o += stride) {
    int m = (int)(o / D_), n = (int)(o % D_);
    float acc = bias[n];
    for (int k = 0; k < D_; ++k) acc += z[(size_t)m * D_ + k] * W[(size_t)k * D_ + n];
    proj[o] = acc;
  }
}
#endif

// ---------------------------------------------------------------------------
// Kernel B: per (b,p): scaled distances to S=1024 sensors, top-32 smallest,
// masked softmax -> weights + indices.  One wave32 per (b,p); each lane holds
// 32 distances in registers; selection via bitmask + shfl_xor butterfly argmin.
// ---------------------------------------------------------------------------
__global__ __launch_bounds__(256) void k_topk_weights(
    const float* __restrict__ Y, const float* __restrict__ coords,
    const unsigned char* __restrict__ mask, const float* __restrict__ phi_mean,
    float* __restrict__ w_buf, int* __restrict__ idx_buf)
{
  const int lane = threadIdx.x & 31;
  const int task = blockIdx.x * (blockDim.x >> 5) + (threadIdx.x >> 5);
  const int b = task / P_;
  const int p = task - b * P_;

  const float yx = Y[((size_t)b * P_ + p) * 2 + 0];
  const float yy = Y[((size_t)b * P_ + p) * 2 + 1];

  float dv[32];
#pragma unroll
  for (int i = 0; i < 32; ++i) {
    const int s = i * 32 + lane;                    // coalesced across lanes
    const size_t cs = (size_t)b * S_ + s;
    const float dx = yx - coords[cs * 2 + 0];
    const float dy = yy - coords[cs * 2 + 1];
    const float d  = sqrtf(fmaxf(dx * dx + dy * dy, 1e-12f));
    const float ds = d / (sqrtf(phi_mean[cs]) + 1e-6f);   // GAMMA = 0.5
    const bool valid = mask[(size_t)b * T_ * S_ + s] != 0; // mask[:,0,:]
    dv[i] = valid ? ds : INFINITY;
  }

  unsigned sel = 0u;
  float sel_d = INFINITY;
  int   sel_s = 0;
  for (int k = 0; k < TOPK; ++k) {
    // local min among unselected
    float best = INFINITY; int bi = 0;
#pragma unroll
    for (int i = 0; i < 32; ++i) {
      if (!((sel >> i) & 1u)) {
        const float v = dv[i];
        if (v < best) { best = v; bi = i; }
      }
    }
    int gidx = bi * 32 + lane;
    // wave argmin (value, tie -> smaller index); butterfly leaves result in all lanes
#pragma unroll
    for (int off = 16; off > 0; off >>= 1) {
      const float ov = __shfl_xor(best, off);
      const int   oi = __shfl_xor(gidx, off);
      if (ov < best || (ov == best && oi < gidx)) { best = ov; gidx = oi; }
    }
    if ((gidx & 31) == lane) sel |= 1u << (gidx >> 5);
    if (lane == k) { sel_d = best; sel_s = gidx; }
  }

  // masked softmax over the 32 selected (lane k holds k-th smallest)
  const float d0 = __shfl(sel_d, 0);               // min distance -> max score
  float e = 0.0f;
  if (sel_d < INFINITY && d0 < INFINITY)
    e = expf(-(sel_d - d0) * (1.0f / SIGMA_));
  float sum = e;
#pragma unroll
  for (int off = 16; off > 0; off >>= 1) sum += __shfl_xor(sum, off);
  const float w = e / (sum + 1e-6f);

  w_buf[(size_t)task * TOPK + lane] = w;
  idx_buf[(size_t)task * TOPK + lane] = sel_s;
}

// ---------------------------------------------------------------------------
// Kernel C: h[bt,p,:] = sum_k w[k] * proj[bt, idx_k, :]; RMS-norm over D=256,
// scale by agg_w, write output.  One wave32 per (bt,p): lane owns 8 channels
// (2x v4f); (w, idx) broadcast per-k via __shfl (wave-uniform -> uniform
// branch skips zero-weight gathers).
//
// The 256 MB output stream exceeds the 192 MB L2: store it NON-TEMPORAL
// (TH=NT) so it streams to HBM and does not evict the L2-resident proj
// slices (1 MB per (b,t), reused 8192x) that feed the gather.
// Uses clang ext-vector v4f (HIP float4 is a class -> rejected by the builtin).
// ---------------------------------------------------------------------------
__global__ __launch_bounds__(256) void k_gather_rms(
    const float* __restrict__ proj, const float* __restrict__ w_buf,
    const int* __restrict__ idx_buf, const float* __restrict__ agg_w,
    float* __restrict__ out)
{
  const int lane = threadIdx.x & 31;
  const int row  = blockIdx.x * (blockDim.x >> 5) + (threadIdx.x >> 5); // [0, BT*P)
  const int bt = row / P_;
  const int p  = row - bt * P_;
  const int b  = bt / T_;

  const size_t wbase = ((size_t)b * P_ + p) * TOPK;
  const float wv = w_buf[wbase + lane];
  const int   sv = idx_buf[wbase + lane];

  v4f acc0 = {0.f, 0.f, 0.f, 0.f};
  v4f acc1 = {0.f, 0.f, 0.f, 0.f};
  const v4f* projv = (const v4f*)proj;

#pragma unroll 8
  for (int k = 0; k < TOPK; ++k) {
    const float wk = __shfl(wv, k);        // wave-uniform
    if (wk == 0.0f) continue;              // uniform branch
    const int sk = __shfl(sv, k);
    const v4f* src = projv + ((size_t)bt * S_ + sk) * (D_ / 4);
    const v4f x0 = src[lane * 2 + 0];
    const v4f x1 = src[lane * 2 + 1];
    acc0 += wk * x0;                       // native vector FMA
    acc1 += wk * x1;
  }

  float ss = acc0.x * acc0.x + acc0.y * acc0.y + acc0.z * acc0.z + acc0.w * acc0.w
           + acc1.x * acc1.x + acc1.y * acc1.y + acc1.z * acc1.z + acc1.w * acc1.w;
#pragma unroll
  for (int off = 16; off > 0; off >>= 1) ss += __shfl_xor(ss, off);
  const float scale = rsqrtf(ss * (1.0f / D_) + 1.1920929e-07f);  // eps = f32 eps

  const v4f* gv = (const v4f*)agg_w;
  const v4f g0 = gv[lane * 2 + 0];
  const v4f g1 = gv[lane * 2 + 1];

  const v4f o0 = acc0 * scale * g0;
  const v4f o1 = acc1 * scale * g1;

  v4f* op = (v4f*)(out + ((size_t)bt * P_ + p) * D_);
  __builtin_nontemporal_store(o0, op + lane * 2 + 0);  // global_store_b128 ... nt
  __builtin_nontemporal_store(o1, op + lane * 2 + 1);
}

// ---------------------------------------------------------------------------
extern "C" void kernel_launch(void* const* d_in, const int* in_sizes, int n_in,
                              void* d_out, int out_size, void* d_ws, size_t ws_size,
                              hipStream_t stream) {
  (void)in_sizes; (void)n_in; (void)out_size; (void)ws_size;
  const float*         z      = (const float*)d_in[0];          // (B,T,S,D)
  const float*         Y      = (const float*)d_in[1];          // (B,P,2)
  const float*         coords = (const float*)d_in[2];          // (B,S,2)
  const unsigned char* mask   = (const unsigned char*)d_in[3];  // (B,T,S) bool
  const float*         phi    = (const float*)d_in[4];          // (B,S)
  const float*         W      = (const float*)d_in[5];          // (D,D)
  const float*         b_lat  = (const float*)d_in[6];          // (D,)
  const float*         agg_w  = (const float*)d_in[7];          // (D,)
  float* out = (float*)d_out;                                   // (B*T,P,D) fp32

  char* ws = (char*)d_ws;
  float* proj = (float*)ws;                                     // 32 MB
  float* wbuf = (float*)(ws + (size_t)M_ * D_ * 4);             //  4 MB
  int*   ibuf = (int*)(ws + (size_t)M_ * D_ * 4 + (size_t)B_ * P_ * TOPK * 4);

#if USE_WMMA_F32
  // 2048 M-tiles x 4 N-tiles = 8192 wave-tiles; 8 waves/block
  k_proj_gemm<<<(M_ / 16) * (D_ / 64) / 8, 256, 0, stream>>>(z, W, b_lat, proj);
#else
  k_proj_gemm<<<2048, 256, 0, stream>>>(z, W, b_lat, proj);
#endif
  k_topk_weights<<<(B_ * P_) / 8, 256, 0, stream>>>(Y, coords, mask, phi, wbuf, ibuf);
  k_gather_rms<<<(BT_ * P_) / 8, 256, 0, stream>>>(proj, wbuf, ibuf, agg_w, out);
}